// SphArr_43696997269680
// MI455X (gfx1250) — compile-verified
//
#include <hip/hip_runtime.h>
#include <stdint.h>

// CDNA5 / gfx1250: wave32, WGP, TDM + async LDS<->global data movers.
// Spherical harmonics l=0..3 for 1,048,576 points: HBM-bound streaming op
// (~140 MB @ 23.3 TB/s ~= 6 us). No contraction => no WMMA; optimize the
// data-movement path with CDNA5 TDM load + async store-from-LDS.

typedef __attribute__((ext_vector_type(4))) unsigned int v4u;
typedef __attribute__((ext_vector_type(8))) int         v8i;
typedef __attribute__((ext_vector_type(4))) int         v4i;

#define TPB 256          // 8 waves of 32; one point per thread
#define COLS 257         // LDS column stride (floats) -> bank-conflict-free

__global__ __launch_bounds__(TPB)
void sph_l4_kernel(const float* __restrict__ X, float* __restrict__ Out)
{
    __shared__ float lds_in[TPB * 3];        // 3072 B staged by TDM
    __shared__ float lds_out[32 * COLS];     // [f][point] transposed staging

    const int t  = threadIdx.x;
    const int p0 = blockIdx.x * TPB;

    // ---------------- input: one TDM DMA per block (wave 0 issues) ---------
    if ((threadIdx.x >> 5) == 0) {
        uint64_t ga  = (uint64_t)(uintptr_t)(X + (size_t)p0 * 3);
        uint32_t lin = (uint32_t)(uintptr_t)(&lds_in[0]);   // LDS byte offset
        v4u g0;
        g0[0] = 1u;                       // count=1, user descriptor
        g0[1] = lin;                      // lds_addr
        g0[2] = (uint32_t)ga;             // global_addr[31:0]
        g0[3] = (uint32_t)((ga >> 32) & 0x01FFFFFFu) | (2u << 30); // addr[56:32] | type=2
        v8i g1;
        g1[0] = 0x00020000;               // data_size=2 -> 4-byte elements
        g1[1] = (int)(768u << 16);        // tensor_dim0 = 768 (lo16 in hi half)
        g1[2] = (int)(1u << 16);          // tensor_dim0 hi=0 ; tensor_dim1 = 1
        g1[3] = (int)(768u << 16);        // tensor_dim1 hi=0 ; tile_dim0 = 768
        g1[4] = 1;                        // tile_dim1 = 1, tile_dim2 = 0
        g1[5] = 768;                      // tensor_dim0_stride lo32
        g1[6] = 0;                        // stride hi / dim1_stride lo
        g1[7] = 0;
        v4i g2 = {0, 0, 0, 0};
        v4i g3 = {0, 0, 0, 0};
        v8i g4 = {0, 0, 0, 0, 0, 0, 0, 0};
        __builtin_amdgcn_tensor_load_to_lds(g0, g1, g2, g3, g4, 0);
        __builtin_amdgcn_s_wait_tensorcnt(0);
    }
    __syncthreads();

    // ---------------- compute: closed-form Y_l^m, l=0..3 -------------------
    const float x = lds_in[3 * t + 0];
    const float y = lds_in[3 * t + 1];
    const float z = lds_in[3 * t + 2];
    const float rinv = rsqrtf(x * x + y * y + z * z);
    const float u = x * rinv, v = y * rinv, w = z * rinv;
    const float u2 = u * u, v2 = v * v, w2 = w * w;
    const float q2r = u2 - v2,            q2i = 2.0f * u * v;     // st^2 e^{2i phi}
    const float q3r = u * (u2 - 3.f*v2),  q3i = v * (3.f*u2 - v2);// st^3 e^{3i phi}
    const float p31 = 5.0f * w2 - 1.0f;

    // Condon-Shortley orthonormal constants
    const float C00 = 0.28209479177387814f;  // 1/sqrt(4pi)
    const float N10 = 0.48860251190291992f;  // sqrt(3/4pi)
    const float N11 = 0.34549414947133547f;  // sqrt(3/8pi)
    const float N20 = 0.63078313050504001f;  // sqrt(5/4pi)
    const float K21 = 0.77254840404637906f;  // sqrt(15/8pi)
    const float K22 = 0.38627420202318953f;  // sqrt(15/32pi)
    const float N30 = 0.74635266518023078f;  // sqrt(7/4pi)
    const float K31 = 0.32318018411415065f;  // sqrt(21/64pi)
    const float K32 = 1.02198547643328230f;  // sqrt(105/32pi)
    const float K33 = 0.41722382363278409f;  // sqrt(35/64pi)

    float o[32];
    // l=0
    o[0]  =  C00;              o[1]  = 0.0f;
    // l=1 : m=-1,0,+1
    o[2]  =  N11 * u;          o[3]  = -N11 * v;
    o[4]  =  N10 * w;          o[5]  = 0.0f;
    o[6]  = -N11 * u;          o[7]  = -N11 * v;
    // l=2 : m=-2..+2
    o[8]  =  K22 * q2r;        o[9]  = -K22 * q2i;
    o[10] =  K21 * w * u;      o[11] = -K21 * w * v;
    o[12] =  N20 * (1.5f * w2 - 0.5f);            o[13] = 0.0f;
    o[14] = -K21 * w * u;      o[15] = -K21 * w * v;
    o[16] =  K22 * q2r;        o[17] =  K22 * q2i;
    // l=3 : m=-3..+3
    o[18] =  K33 * q3r;        o[19] = -K33 * q3i;
    o[20] =  K32 * w * q2r;    o[21] = -K32 * w * q2i;
    o[22] =  K31 * p31 * u;    o[23] = -K31 * p31 * v;
    o[24] =  N30 * (2.5f * w2 - 1.5f) * w;        o[25] = 0.0f;
    o[26] = -K31 * p31 * u;    o[27] = -K31 * p31 * v;
    o[28] =  K32 * w * q2r;    o[29] =  K32 * w * q2i;
    o[30] = -K33 * q3r;        o[31] = -K33 * q3i;

    // transpose-stage: lds_out[f][point]; lanes hit consecutive banks (no conflicts)
#pragma unroll
    for (int f = 0; f < 32; ++f) lds_out[f * COLS + t] = o[f];
    __syncthreads();

    // ---------------- output: coalesced async LDS -> global ----------------
    // global float index g = t + 256*k  (block-contiguous per k)
    // point p = g/32 = 8k + t/32 ; component f = g%32 = t&31
    // LDS float addr = f*COLS + p  -> 32 distinct banks per instruction.
    float* outp = Out + (size_t)p0 * 32;
    const uint32_t lobase = (uint32_t)(uintptr_t)(&lds_out[0]);
    const int lane = t & 31;
    const int wv   = t >> 5;
#pragma unroll 4
    for (int k = 0; k < 32; ++k) {
        uint32_t ldsb = lobase + 4u * (uint32_t)(lane * COLS + 8 * k + wv);
        uint32_t goff = 4u * (uint32_t)(t + TPB * k);
        asm volatile("global_store_async_from_lds_b32 %0, %1, %2"
                     :: "v"(goff), "v"(ldsb), "s"(outp)
                     : "memory");
    }
    asm volatile("s_wait_asynccnt 0x0" ::: "memory");
}

extern "C" void kernel_launch(void* const* d_in, const int* in_sizes, int n_in,
                              void* d_out, int out_size, void* d_ws, size_t ws_size,
                              hipStream_t stream)
{
    (void)n_in; (void)out_size; (void)d_ws; (void)ws_size;
    const float* X = (const float*)d_in[0];
    float* Out = (float*)d_out;
    const int npts = in_sizes[0] / 3;            // B*N*N points
    const int nblocks = (npts + TPB - 1) / TPB;  // exact: 4096
    sph_l4_kernel<<<dim3(nblocks), dim3(TPB), 0, stream>>>(X, Out);
}